// HiPPOLegSKernel_64381559767205
// MI455X (gfx1250) — compile-verified
//
#include <hip/hip_runtime.h>

typedef __attribute__((ext_vector_type(2))) float v2f;
typedef __attribute__((ext_vector_type(8))) float v8f;

constexpr int SN    = 64;      // state dim
constexpr int DM    = 1024;    // d_model
constexpr int SEQL  = 2048;    // sequence length
constexpr int MAT   = SN * SN; // 4096 floats per 64x64 matrix

// workspace layout (float offsets)
constexpr long ABAR = 0;                      // 64x64 A_bar (row major)
constexpr long LINV = ABAR + MAT;             // 64x64 lhs^-1 (row major)
constexpr long A64O = LINV + MAT;             // 64x64 A_bar^64
constexpr long DTV  = A64O + MAT;             // dt scalar (padded slot)
constexpr long PBUF = DTV + SN;               // 64 mats: A_bar^j, j=0..63
constexpr long QBUF = PBUF + 64L * MAT;       // 32 mats: (A_bar^64)^i, i=0..31
constexpr long GBUF = QBUF + 32L * MAT;       // 2048 x 4096: G[l][n*64+m] = (A_bar^l)[n][m]
constexpr long OBUF = GBUF + (long)SEQL*MAT;  // 1024 x 4096: O[d][n*64+m] = C[d][n]*Bbar[d][m]
// total = OBUF + 1024*4096 floats  ~= 49.6 MB

__device__ __forceinline__ float hippoA(int i, int j) {
    if (j < i)  return -sqrtf((2.0f*i + 1.0f) * (2.0f*j + 1.0f));
    if (j == i) return -(float)(i + 1);
    return 0.0f;
}

// ---------------------------------------------------------------------------
// Kernel 0: discretize.  lhs = I - dt/2 A is LOWER TRIANGULAR with diag >= 1,
// so solve lhs * X = [rhs | I] by forward substitution, one column per thread.
// Writes A_bar, Linv = lhs^-1, dt.
// ---------------------------------------------------------------------------
__global__ void setup_kernel(const float* __restrict__ log_dt, float* __restrict__ ws) {
    __shared__ float lhsS[SN][SN];        // 16 KB
    __shared__ float xS[SN][128];         // 32 KB, column per thread
    const int t = threadIdx.x;            // 0..127
    const float dt = log1pf(expf(log_dt[0])) + 1e-6f;

    for (int e = t; e < MAT; e += 128) {
        int i = e >> 6, j = e & 63;
        lhsS[i][j] = (i == j ? 1.0f : 0.0f) - 0.5f * dt * hippoA(i, j);
    }
    __syncthreads();

    const int  c     = t & 63;
    const bool isRhs = (t < 64);          // first 64 threads: columns of rhs -> A_bar
    for (int i = 0; i < SN; ++i) {
        float bi;
        if (isRhs) bi = (i == c ? 1.0f : 0.0f) + 0.5f * dt * hippoA(i, c);
        else       bi = (i == c ? 1.0f : 0.0f);
        float s = bi;
        for (int j = 0; j < i; ++j) s -= lhsS[i][j] * xS[j][t];
        xS[i][t] = s / lhsS[i][i];
    }
    float* dst = ws + (isRhs ? ABAR : LINV);
    for (int i = 0; i < SN; ++i) dst[i*SN + c] = xS[i][t];
    if (t == 0) ws[DTV] = dt;
}

// ---------------------------------------------------------------------------
// Kernel 1: sequential power chains (single block).
//   P_j = A_bar^j  (j=0..63),  A64 = A_bar^64,  Q_i = A64^i (i=0..31)
// ---------------------------------------------------------------------------
__global__ void power_kernel(float* __restrict__ ws) {
    __shared__ float Mul[SN][SN];
    __shared__ float Cur[SN][SN];
    __shared__ float Nxt[SN][SN];
    const int t = threadIdx.x;            // 0..255

    for (int e = t; e < MAT; e += 256) {
        Mul[e>>6][e&63] = ws[ABAR + e];
        Cur[e>>6][e&63] = ((e>>6) == (e&63)) ? 1.0f : 0.0f;
    }
    __syncthreads();

    for (int j = 0; j < 64; ++j) {
        for (int e = t; e < MAT; e += 256) ws[PBUF + (long)j*MAT + e] = Cur[e>>6][e&63];
        for (int e = t; e < MAT; e += 256) {
            int i = e >> 6, m = e & 63;
            float s = 0.0f;
            for (int k = 0; k < SN; ++k) s += Cur[i][k] * Mul[k][m];
            Nxt[i][m] = s;
        }
        __syncthreads();
        for (int e = t; e < MAT; e += 256) Cur[e>>6][e&63] = Nxt[e>>6][e&63];
        __syncthreads();
    }
    for (int e = t; e < MAT; e += 256) ws[A64O + e] = Cur[e>>6][e&63];
    __syncthreads();
    for (int e = t; e < MAT; e += 256) {
        Mul[e>>6][e&63] = Cur[e>>6][e&63];            // multiplier = A_bar^64
        Cur[e>>6][e&63] = ((e>>6) == (e&63)) ? 1.0f : 0.0f;
    }
    __syncthreads();
    for (int i2 = 0; i2 < 32; ++i2) {
        for (int e = t; e < MAT; e += 256) ws[QBUF + (long)i2*MAT + e] = Cur[e>>6][e&63];
        for (int e = t; e < MAT; e += 256) {
            int i = e >> 6, m = e & 63;
            float s = 0.0f;
            for (int k = 0; k < SN; ++k) s += Cur[i][k] * Mul[k][m];
            Nxt[i][m] = s;
        }
        __syncthreads();
        for (int e = t; e < MAT; e += 256) Cur[e>>6][e&63] = Nxt[e>>6][e&63];
        __syncthreads();
    }
}

// ---------------------------------------------------------------------------
// Kernel 2: G_l = Q_{l>>6} @ P_{l&63}, one block per l (fully parallel).
// ---------------------------------------------------------------------------
__global__ void gl_kernel(float* __restrict__ ws) {
    const int l = blockIdx.x;
    __shared__ float Qs[SN][SN], Ps[SN][SN];
    const int t = threadIdx.x;            // 0..255
    const float* Qg = ws + QBUF + (long)(l >> 6) * MAT;
    const float* Pg = ws + PBUF + (long)(l & 63) * MAT;
    for (int e = t; e < MAT; e += 256) { Qs[e>>6][e&63] = Qg[e]; Ps[e>>6][e&63] = Pg[e]; }
    __syncthreads();
    float* G = ws + GBUF + (long)l * MAT;
    for (int e = t; e < MAT; e += 256) {
        int n = e >> 6, m = e & 63;
        float s = 0.0f;
        for (int k = 0; k < SN; ++k) s += Qs[n][k] * Ps[k][m];
        G[e] = s;
    }
}

// ---------------------------------------------------------------------------
// Kernel 3: Bbar[d,m] = dt * sum_k B[d,k]*Linv[m,k];  O[d, n*64+m] = C[d,n]*Bbar[d,m]
// ---------------------------------------------------------------------------
__global__ void o_kernel(const float* __restrict__ Bmat, const float* __restrict__ Cmat,
                         float* __restrict__ ws) {
    const int d = blockIdx.x;
    const int t = threadIdx.x;            // 0..63
    __shared__ float Brow[SN], Crow[SN], Bbar[SN];
    const float dt = ws[DTV];
    Brow[t] = Bmat[d*SN + t];
    Crow[t] = Cmat[d*SN + t];
    __syncthreads();
    const float* Linv = ws + LINV;
    float s = 0.0f;
    for (int k = 0; k < SN; ++k) s += Brow[k] * Linv[t*SN + k];
    Bbar[t] = dt * s;
    __syncthreads();
    float* O = ws + OBUF + (long)d * MAT;
    const float bm = Bbar[t];
    for (int n = 0; n < SN; ++n) O[n*SN + t] = Crow[n] * bm;
}

// ---------------------------------------------------------------------------
// Kernel 4: K = O (1024x4096) * G^T (4096x2048) via V_WMMA_F32_16X16X4_F32.
// One wave per 16(M) x 64(N) tile: 4 accumulators, A-frag reused 4x per k-step.
// f32 A 16x4 layout: lane = M row (lo), K split by lane half (2*hi + {0,1}).
// f32 B 4x16 layout: lane = N col (lo), same K split -> contiguous float2 loads.
// ---------------------------------------------------------------------------
__global__ void gemm_kernel(const float* __restrict__ O, const float* __restrict__ G,
                            const float* __restrict__ Dvec, float* __restrict__ out) {
    const int m0   = blockIdx.x * 16;
    const int n0   = blockIdx.y * 64;
    const int lane = threadIdx.x;         // 0..31, full wave, no divergence
    const int lo   = lane & 15;
    const int hi   = lane >> 4;

    const float* arow = O + (long)(m0 + lo) * MAT + 2*hi;
    const float* brow = G + (long)(n0 + lo) * MAT + 2*hi;

    v8f c0 = {}, c1 = {}, c2 = {}, c3 = {};
    for (int k = 0; k < MAT; k += 4) {
        v2f a  = *(const v2f*)(arow + k);
        v2f b0 = *(const v2f*)(brow + k);
        v2f b1 = *(const v2f*)(brow + 16L*MAT + k);
        v2f b2 = *(const v2f*)(brow + 32L*MAT + k);
        v2f b3 = *(const v2f*)(brow + 48L*MAT + k);
        c0 = __builtin_amdgcn_wmma_f32_16x16x4_f32(false, a, false, b0, (short)0, c0, false, false);
        c1 = __builtin_amdgcn_wmma_f32_16x16x4_f32(false, a, false, b1, (short)0, c1, false, false);
        c2 = __builtin_amdgcn_wmma_f32_16x16x4_f32(false, a, false, b2, (short)0, c2, false, false);
        c3 = __builtin_amdgcn_wmma_f32_16x16x4_f32(false, a, false, b3, (short)0, c3, false, false);
    }

    // D-matrix layout: VGPR v holds row M = 8*hi + v, column N = lo.
    #pragma unroll
    for (int v = 0; v < 8; ++v) {
        const int dd = m0 + 8*hi + v;
        float* orow = out + (long)dd * SEQL + n0 + lo;
        const float add0 = (n0 + lo == 0) ? Dvec[dd] : 0.0f;  // K[:,0] += D
        orow[0]  = c0[v] + add0;
        orow[16] = c1[v];
        orow[32] = c2[v];
        orow[48] = c3[v];
    }
}

extern "C" void kernel_launch(void* const* d_in, const int* in_sizes, int n_in,
                              void* d_out, int out_size, void* d_ws, size_t ws_size,
                              hipStream_t stream) {
    const float* B      = (const float*)d_in[0];   // (1024, 64)
    const float* C      = (const float*)d_in[1];   // (1024, 64)
    const float* D      = (const float*)d_in[2];   // (1024,)
    const float* log_dt = (const float*)d_in[3];   // scalar
    float* ws  = (float*)d_ws;
    float* out = (float*)d_out;                    // (1024, 1, 2048) f32

    setup_kernel<<<1, 128, 0, stream>>>(log_dt, ws);
    power_kernel<<<1, 256, 0, stream>>>(ws);
    gl_kernel<<<SEQL, 256, 0, stream>>>(ws);
    o_kernel<<<DM, SN, 0, stream>>>(B, C, ws);
    gemm_kernel<<<dim3(DM/16, SEQL/64), 32, 0, stream>>>(ws + OBUF, ws + GBUF, D, out);
}